// _WeightedDeltaParametrization_8890582303285
// MI455X (gfx1250) — compile-verified
//
#include <hip/hip_runtime.h>

// out = base + sum_i coeffs[i] * deltas[i]   (4096x4096 fp32, 8 deltas)
//
// Bandwidth-bound (640 MiB / call, ~0.4 FLOP/B -> ~28us floor at 23.3 TB/s).
// The delta-axis contraction is mapped onto V_WMMA_F32_16X16X4_F32:
//   A(16x4)  = coeffs broadcast down rows   (VGPR0 = K0|K2, VGPR1 = K1|K3)
//   B(4x16)  = deltas[k][j+n] loaded straight into the B lane layout
//   C        = inline 0 (compiler emits "v_wmma ..., 0" -> no C register
//              materialization); base is added full-wave AFTER the tile merge.
// Two chained WMMAs cover k=0..7; D row 0 (VGPR0, lanes 0-15) is the tile sum.
// Two adjacent 16-column tiles per wave iteration + one ds_swizzle SWAPX16
// give full-wave base loads and stores. All traffic uses non-temporal hints
// (zero reuse; 640 MB working set >> 192 MB L2).

typedef __attribute__((ext_vector_type(2))) float v2f;
typedef __attribute__((ext_vector_type(8))) float v8f;

static constexpr unsigned kPlane = 4096u * 4096u;  // elements per plane
#define NT_LOAD(p) __builtin_nontemporal_load(p)

__global__ __launch_bounds__(256) void wdelta_wmma_kernel(
    const float* __restrict__ base,
    const float* __restrict__ coeffs,
    const float* __restrict__ deltas,
    float* __restrict__ out)
{
  const unsigned lane = threadIdx.x & 31u;
  const bool     hi   = lane >= 16u;        // lane half selects K row pair
  const unsigned n    = lane & 15u;         // column within the 16-wide tile

  const unsigned wavesPerBlock = blockDim.x >> 5;
  const unsigned waveId  = blockIdx.x * wavesPerBlock + (threadIdx.x >> 5);
  const unsigned nWaves  = gridDim.x * wavesPerBlock;
  const unsigned jStride = nWaves * 32u;    // 32 outputs per wave iteration

  // coeffs are wave-uniform -> scalar loads
  const float c0 = coeffs[0], c1 = coeffs[1], c2 = coeffs[2], c3 = coeffs[3];
  const float c4 = coeffs[4], c5 = coeffs[5], c6 = coeffs[6], c7 = coeffs[7];

  // A operand (16x4 f32), A[m][k] = coeffs[k] independent of m.
  // ISA layout: VGPR0 = K=0 (lanes 0-15) / K=2 (lanes 16-31); VGPR1 = K=1 / K=3.
  v2f aLo, aHi4;
  aLo[0]  = hi ? c2 : c0;
  aLo[1]  = hi ? c3 : c1;
  aHi4[0] = hi ? c6 : c4;
  aHi4[1] = hi ? c7 : c5;

  // Per-lane plane offsets so plain vector loads land in the B(4x16) layout:
  // VGPR0 holds row k (lanes 0-15) / k+2 (lanes 16-31); VGPR1 holds k+1 / k+3.
  const unsigned pLo = (hi ? 2u : 0u) * kPlane;
  const unsigned pHi = (hi ? 3u : 1u) * kPlane;
  const unsigned p4  = 4u * kPlane;

  const v8f zeroC = {0.f, 0.f, 0.f, 0.f, 0.f, 0.f, 0.f, 0.f};

  for (unsigned j = waveId * 32u; j < kPlane; j += jStride) {
    const unsigned colA = j + n;          // tile A: columns j .. j+15
    const unsigned colB = colA + 16u;     // tile B: columns j+16 .. j+31

    // B operands (non-temporal: streamed exactly once)
    v2f bAlo = { NT_LOAD(deltas + pLo + colA),      NT_LOAD(deltas + pHi + colA)      };
    v2f bAhi = { NT_LOAD(deltas + p4 + pLo + colA), NT_LOAD(deltas + p4 + pHi + colA) };
    v2f bBlo = { NT_LOAD(deltas + pLo + colB),      NT_LOAD(deltas + pHi + colB)      };
    v2f bBhi = { NT_LOAD(deltas + p4 + pLo + colB), NT_LOAD(deltas + p4 + pHi + colB) };

    // Full-wave base load: lane L holds base[j+L].
    const float bv = NT_LOAD(base + j + lane);

    // D = A x B + 0, chained over the two K=4 halves (k=0..3 then k=4..7).
    v8f dA = __builtin_amdgcn_wmma_f32_16x16x4_f32(false, aLo,  false, bAlo,
                                                   (short)0, zeroC, false, false);
    dA     = __builtin_amdgcn_wmma_f32_16x16x4_f32(false, aHi4, false, bAhi,
                                                   (short)0, dA, false, false);
    v8f dB = __builtin_amdgcn_wmma_f32_16x16x4_f32(false, aLo,  false, bBlo,
                                                   (short)0, zeroC, false, false);
    dB     = __builtin_amdgcn_wmma_f32_16x16x4_f32(false, aHi4, false, bBhi,
                                                   (short)0, dB, false, false);

    // D row 0 lives in VGPR0, lanes 0-15 of each tile. Route tile B's sums
    // into lanes 16-31 (SWAPX16: xor=0x10, and=0x1f), merge, add base once
    // full-wave, then a single full-wave NT store.
    const float sA = dA[0];
    const float sBsw =
        __int_as_float(__builtin_amdgcn_ds_swizzle(__float_as_int(dB[0]), 0x401F));
    const float r = (hi ? sBsw : sA) + bv;
    __builtin_nontemporal_store(r, out + j + lane);
  }
}

extern "C" void kernel_launch(void* const* d_in, const int* in_sizes, int n_in,
                              void* d_out, int out_size, void* d_ws, size_t ws_size,
                              hipStream_t stream) {
    const float* base   = (const float*)d_in[0];   // (4096, 4096) f32
    const float* coeffs = (const float*)d_in[1];   // (8,) f32
    const float* deltas = (const float*)d_in[2];   // (8, 4096, 4096) f32
    float* out = (float*)d_out;                    // (4096, 4096) f32

    // 4096 blocks x 256 threads = 32768 waves; each wave streams 32 outputs
    // per iteration, 16 grid-stride iterations total. Plenty of memory-level
    // parallelism to keep 23.3 TB/s of HBM busy.
    dim3 grid(4096), block(256);
    hipLaunchKernelGGL(wdelta_wmma_kernel, grid, block, 0, stream,
                       base, coeffs, deltas, out);
}